// GATLayer_18064632447086
// MI455X (gfx1250) — compile-verified
//
#include <hip/hip_runtime.h>
#include <math.h>

#define HEADS 4
#define CPH   16
#define HC    64      // HEADS*CPH
#define EDIM  16
#define NSLOPE 0.2f

typedef float v2f __attribute__((ext_vector_type(2)));
typedef float v8f __attribute__((ext_vector_type(8)));

// ---- monotone uint key for f32 atomic max (handles negatives) ----
__device__ __forceinline__ unsigned fkey(float x) {
  unsigned b = __float_as_uint(x);
  return (b & 0x80000000u) ? ~b : (b | 0x80000000u);
}
__device__ __forceinline__ float funkey(unsigned k) {
  unsigned b = (k & 0x80000000u) ? (k & 0x7FFFFFFFu) : ~k;
  return __uint_as_float(b);
}

// ---------------------------------------------------------------------------
// h = x @ W  via V_WMMA_F32_16X16X4_F32.
// Block = 128 threads (4 waves). Block b owns M-tile b (16 rows); wave w owns
// N-tile w (16 of the 64 output columns). K loop: IN/4 WMMA steps.
// A frag (16x4 f32, ISA 7.12.2): lanes 0-15 -> (M=lane, K=k0..k0+1),
//                                lanes 16-31 -> (M=lane-16, K=k0+2..k0+3).
// B frag (4x16 f32): lanes 0-15 -> (N=lane, K=k0..k0+1), lanes16-31 -> K+2.
// C/D (16x16 f32): VGPR j: lanes0-15 = (M=j, N=lane); lanes16-31 = (M=j+8).
// ---------------------------------------------------------------------------
__global__ void gat_gemm_h(const float* __restrict__ x, const float* __restrict__ W,
                           float* __restrict__ h, int N, int K) {
  const int wave = threadIdx.x >> 5;
  const int lane = threadIdx.x & 31;
  const int half = lane >> 4;     // 0: K pair 0/1, 1: K pair 2/3
  const int l16  = lane & 15;
  const int mbase = blockIdx.x * 16;
  const int nbase = wave * 16;

  int arow = mbase + l16;
  if (arow >= N) arow = N - 1;    // clamp (no EXEC divergence; dup reads ok)
  const float* __restrict__ xr = x + (size_t)arow * K;
  const int ncol = nbase + l16;

  v8f acc = {0.f, 0.f, 0.f, 0.f, 0.f, 0.f, 0.f, 0.f};
  for (int k0 = 0; k0 < K; k0 += 4) {
    const int ka = k0 + 2 * half;
    v2f a, b;
    a.x = xr[ka];
    a.y = xr[ka + 1];
    b.x = W[(size_t)ka * HC + ncol];
    b.y = W[(size_t)(ka + 1) * HC + ncol];
    acc = __builtin_amdgcn_wmma_f32_16x16x4_f32(false, a, false, b,
                                                (short)0, acc, false, false);
  }
#pragma unroll
  for (int j = 0; j < 8; ++j) {
    int m = mbase + j + 8 * half;
    if (m < N) h[(size_t)m * HC + ncol] = acc[j];
  }
}

// ---------------------------------------------------------------------------
// Per-node attention logits: a_src[n,h] = h[n,h,:].att_src[h,:], same for dst.
// ---------------------------------------------------------------------------
__global__ void gat_node_attn(const float* __restrict__ h,
                              const float* __restrict__ att_src,
                              const float* __restrict__ att_dst,
                              float* __restrict__ a_src, float* __restrict__ a_dst,
                              int N) {
  __shared__ float s_att[2 * HC];
  if (threadIdx.x < 2 * HC)
    s_att[threadIdx.x] = (threadIdx.x < HC) ? att_src[threadIdx.x]
                                            : att_dst[threadIdx.x - HC];
  __syncthreads();
  int n = blockIdx.x * blockDim.x + threadIdx.x;
  if (n >= N) return;
  const float* hr = h + (size_t)n * HC;
#pragma unroll
  for (int hh = 0; hh < HEADS; ++hh) {
    float s = 0.f, d = 0.f;
#pragma unroll
    for (int c = 0; c < CPH; ++c) {
      float v = hr[hh * CPH + c];
      s += v * s_att[hh * CPH + c];
      d += v * s_att[HC + hh * CPH + c];
    }
    a_src[(size_t)n * HEADS + hh] = s;
    a_dst[(size_t)n * HEADS + hh] = d;
  }
}

// ---------------------------------------------------------------------------
// Fold W_edge & att_edge: M[d,h] = sum_c W_edge[d, h*16+c]*att_edge[h,c],
// Msum[h] = sum_d M[d,h] (a_e for self-loop edges whose attrs are all 1.0).
// ---------------------------------------------------------------------------
__global__ void gat_edge_fold(const float* __restrict__ W_edge,
                              const float* __restrict__ att_edge,
                              float* __restrict__ Mbuf /* [EDIM*HEADS + HEADS] */) {
  __shared__ float sM[EDIM * HEADS];
  int t = threadIdx.x;
  if (t < EDIM * HEADS) {
    int d = t / HEADS, hh = t % HEADS;
    float s = 0.f;
    for (int c = 0; c < CPH; ++c)
      s += W_edge[(size_t)d * HC + hh * CPH + c] * att_edge[hh * CPH + c];
    sM[t] = s;
    Mbuf[t] = s;
  }
  __syncthreads();
  if (t < HEADS) {
    float s = 0.f;
    for (int d = 0; d < EDIM; ++d) s += sM[d * HEADS + t];
    Mbuf[EDIM * HEADS + t] = s;
  }
}

// ---------------------------------------------------------------------------
// Init: out = bias broadcast; amax keys = 0 (== -inf under fkey order); denom=0.
// ---------------------------------------------------------------------------
__global__ void gat_init(float* __restrict__ out, const float* __restrict__ bias,
                         unsigned* __restrict__ amaxk, float* __restrict__ denom,
                         int N) {
  int i = blockIdx.x * blockDim.x + threadIdx.x;
  if (i < N * HC) out[i] = bias[i & (HC - 1)];
  if (i < N * HEADS) { amaxk[i] = 0u; denom[i] = 0.f; }
}

// ---------------------------------------------------------------------------
// Per-edge logits (leaky-relu'd) + running segment max over dst via atomicMax.
// Edges [0,E) are real; [E, E+N) are self-loops with edge_attr == 1.0.
// ---------------------------------------------------------------------------
__global__ void gat_edge_logits(const int* __restrict__ ei,
                                const float* __restrict__ ea,
                                const float* __restrict__ a_src,
                                const float* __restrict__ a_dst,
                                const float* __restrict__ Mbuf,
                                float* __restrict__ alpha,
                                unsigned* __restrict__ amaxk,
                                int E, int Etot) {
  __shared__ float sM[EDIM * HEADS + HEADS];
  if (threadIdx.x < EDIM * HEADS + HEADS) sM[threadIdx.x] = Mbuf[threadIdx.x];
  __syncthreads();
  int e = blockIdx.x * blockDim.x + threadIdx.x;
  if (e >= Etot) return;

  int s, d;
  float ae[HEADS];
  if (e < E) {
    s = ei[e];
    d = ei[(size_t)E + e];
#pragma unroll
    for (int hh = 0; hh < HEADS; ++hh) ae[hh] = 0.f;
    for (int dd = 0; dd < EDIM; ++dd) {
      float v = ea[(size_t)e * EDIM + dd];
#pragma unroll
      for (int hh = 0; hh < HEADS; ++hh) ae[hh] += v * sM[dd * HEADS + hh];
    }
  } else {
    s = d = e - E;
#pragma unroll
    for (int hh = 0; hh < HEADS; ++hh) ae[hh] = sM[EDIM * HEADS + hh];
  }
#pragma unroll
  for (int hh = 0; hh < HEADS; ++hh) {
    float al = a_src[(size_t)s * HEADS + hh] + a_dst[(size_t)d * HEADS + hh] + ae[hh];
    al = (al > 0.f) ? al : NSLOPE * al;
    alpha[(size_t)e * HEADS + hh] = al;
    atomicMax(&amaxk[(size_t)d * HEADS + hh], fkey(al));
  }
}

// ---------------------------------------------------------------------------
// ex = exp(alpha - amax[dst]) (stored over alpha); denom[dst] += ex.
// ---------------------------------------------------------------------------
__global__ void gat_edge_exp(const int* __restrict__ ei,
                             const unsigned* __restrict__ amaxk,
                             float* __restrict__ alpha,
                             float* __restrict__ denom,
                             int E, int Etot) {
  int e = blockIdx.x * blockDim.x + threadIdx.x;
  if (e >= Etot) return;
  int d = (e < E) ? ei[(size_t)E + e] : (e - E);
#pragma unroll
  for (int hh = 0; hh < HEADS; ++hh) {
    float mx = funkey(amaxk[(size_t)d * HEADS + hh]);
    float ex = expf(alpha[(size_t)e * HEADS + hh] - mx);
    alpha[(size_t)e * HEADS + hh] = ex;
    atomicAdd(&denom[(size_t)d * HEADS + hh], ex);
  }
}

// ---------------------------------------------------------------------------
// out[dst, t] += (ex/denom[dst,h]) * h[src, t]; 64 threads per edge.
// ---------------------------------------------------------------------------
__global__ void gat_aggregate(const int* __restrict__ ei,
                              const float* __restrict__ h,
                              const float* __restrict__ ex,
                              const float* __restrict__ denom,
                              float* __restrict__ out,
                              int E, long long total) {
  long long tid = (long long)blockIdx.x * blockDim.x + threadIdx.x;
  if (tid >= total) return;
  int e  = (int)(tid >> 6);
  int t  = (int)(tid & 63);
  int hh = t >> 4;
  int s, d;
  if (e < E) { s = ei[e]; d = ei[(size_t)E + e]; }
  else       { s = d = e - E; }
  float att = ex[(size_t)e * HEADS + hh] / denom[(size_t)d * HEADS + hh];
  atomicAdd(&out[(size_t)d * HC + t], att * h[(size_t)s * HC + t]);
}

// ---------------------------------------------------------------------------
extern "C" void kernel_launch(void* const* d_in, const int* in_sizes, int n_in,
                              void* d_out, int out_size, void* d_ws, size_t ws_size,
                              hipStream_t stream) {
  const float* x        = (const float*)d_in[0];
  const int*   ei       = (const int*)  d_in[1];
  const float* ea       = (const float*)d_in[2];
  const float* W        = (const float*)d_in[3];
  const float* att_src  = (const float*)d_in[4];
  const float* att_dst  = (const float*)d_in[5];
  const float* W_edge   = (const float*)d_in[6];
  const float* att_edge = (const float*)d_in[7];
  const float* bias     = (const float*)d_in[8];

  const int IN   = in_sizes[3] / HC;     // 128
  const int N    = in_sizes[0] / IN;     // 50000
  const int E    = in_sizes[1] / 2;      // 1.6M
  const int Etot = E + N;                // with self loops

  // ---- carve workspace ----
  char* ws = (char*)d_ws;
  size_t off = 0;
  auto carve = [&](size_t bytes) -> char* {
    char* p = ws + off;
    off = (off + bytes + 255) & ~(size_t)255;
    return p;
  };
  float*    h      = (float*)   carve((size_t)N * HC * sizeof(float));
  float*    a_src  = (float*)   carve((size_t)N * HEADS * sizeof(float));
  float*    a_dst  = (float*)   carve((size_t)N * HEADS * sizeof(float));
  float*    Mbuf   = (float*)   carve((EDIM * HEADS + HEADS) * sizeof(float));
  unsigned* amaxk  = (unsigned*)carve((size_t)N * HEADS * sizeof(unsigned));
  float*    denom  = (float*)   carve((size_t)N * HEADS * sizeof(float));
  float*    alpha  = (float*)   carve((size_t)Etot * HEADS * sizeof(float));
  (void)ws_size; (void)n_in; (void)out_size;
  float* out = (float*)d_out;

  // 1) h = x @ W via WMMA (128 thr = 4 waves, each wave one 16x16 N-tile)
  gat_gemm_h<<<dim3((N + 15) / 16), dim3(128), 0, stream>>>(x, W, h, N, IN);
  // 2) per-node attention dots
  gat_node_attn<<<dim3((N + 255) / 256), dim3(256), 0, stream>>>(
      h, att_src, att_dst, a_src, a_dst, N);
  // 3) fold edge weights (tiny)
  gat_edge_fold<<<dim3(1), dim3(64), 0, stream>>>(W_edge, att_edge, Mbuf);
  // 4) init out/bias + softmax state
  gat_init<<<dim3(((size_t)N * HC + 255) / 256), dim3(256), 0, stream>>>(
      out, bias, amaxk, denom, N);
  // 5) per-edge logits + segment max
  gat_edge_logits<<<dim3((Etot + 255) / 256), dim3(256), 0, stream>>>(
      ei, ea, a_src, a_dst, Mbuf, alpha, amaxk, E, Etot);
  // 6) exp + denominator
  gat_edge_exp<<<dim3((Etot + 255) / 256), dim3(256), 0, stream>>>(
      ei, amaxk, alpha, denom, E, Etot);
  // 7) weighted scatter-add aggregation
  long long total = (long long)Etot * HC;
  gat_aggregate<<<dim3((unsigned)((total + 255) / 256)), dim3(256), 0, stream>>>(
      ei, h, alpha, denom, out, E, total);
}